// HydrogenPool_24137716203759
// MI455X (gfx1250) — compile-verified
//
#include <hip/hip_runtime.h>
#include <cstdint>

typedef __attribute__((ext_vector_type(2))) float v2f;
typedef __attribute__((ext_vector_type(8))) float v8f;

#define BOHR_INV 0.5f   // 1/BOHR, BOHR = MAX_SCALE/N_MAX = 2.0

// ---------------------------------------------------------------------------
// Kernel 0: pack {x,y,z (in Bohr units), f} into contiguous float4 stream in
// workspace so the main kernel can async-DMA B128 tiles straight into LDS.
// ---------------------------------------------------------------------------
__global__ void pack_pts_kernel(const float* __restrict__ f,
                                const float* __restrict__ coords,
                                float4* __restrict__ pts, int total) {
  int i = blockIdx.x * blockDim.x + threadIdx.x;
  if (i < total) {
    float x = coords[3 * i + 0] * BOHR_INV;
    float y = coords[3 * i + 1] * BOHR_INV;
    float z = coords[3 * i + 2] * BOHR_INV;
    pts[i] = make_float4(x, y, z, f[i]);
  }
}

// ---------------------------------------------------------------------------
// Main kernel: one wave (32 lanes) per block, 16 output points per wave.
// grid.x = B*A/16 tiles, grid.y = NSPLIT partial sums over the n dimension.
// Pairwise r^2 for a 16n x 16a tile via V_WMMA_F32_16X16X4_F32:
//   A[n, 0..3] = (x_n, y_n, z_n, |ri_n|^2),  B[0..3, a] = (-2Xa, -2Ya, -2Za, 1)
//   C[n, a]    = |Ro_a|^2     =>  D[n, a] = r^2(a, n)
// ---------------------------------------------------------------------------
__launch_bounds__(32)
__global__ void hydrogen_pool_kernel(const float* __restrict__ out_coords,
                                     const float4* __restrict__ pts,
                                     float* __restrict__ out,
                                     int A, int N, int nPerSplit) {
  __shared__ float4 tile4[2][64];   // double-buffered 64-point tiles (2 KB)

  const int lane = threadIdx.x;     // wave32
  const int col  = lane & 15;       // output point within tile (WMMA column)
  const bool hi  = (lane & 16) != 0; // which half of the 16 rows this lane owns

  const int atiles = A / 16;
  const int b  = blockIdx.x / atiles;
  const int at = blockIdx.x % atiles;
  const int a  = at * 16 + col;

  // Output point coords (Bohr units) — lanes l and l+16 share the same a.
  const float Xa = out_coords[((size_t)(b * A + a)) * 3 + 0] * BOHR_INV;
  const float Ya = out_coords[((size_t)(b * A + a)) * 3 + 1] * BOHR_INV;
  const float Za = out_coords[((size_t)(b * A + a)) * 3 + 2] * BOHR_INV;
  const float q  = Xa * Xa + Ya * Ya + Za * Za;

  // B matrix regs: lanes 0-15 hold rows K0,K1; lanes 16-31 rows K2,K3.
  // Branchless (v_cndmask), so no exec-mask diamonds anywhere near the WMMA.
  v2f Bm;
  Bm.x = hi ? -2.0f * Za : -2.0f * Xa;
  Bm.y = hi ? 1.0f       : -2.0f * Ya;

  // C matrix: every row of this lane's column is |Ro_a|^2.
  v8f Cm;
#pragma unroll
  for (int j = 0; j < 8; ++j) Cm[j] = q;

  const int nbase = b * N + blockIdx.y * nPerSplit;
  const int nChunks = nPerSplit / 64;

  const unsigned ldsBase =
      (unsigned)(uintptr_t)&tile4[0][0];          // flat LDS addr low 32 bits
  const unsigned long long pbase = (unsigned long long)(uintptr_t)pts;

  // Async DMA one 64-point (1 KB) tile into LDS buffer `buf`.
  // Each lane issues 2x B128; inst offset:512 advances BOTH LDS and global
  // sides per the async addressing rules, so one reg pair serves both copies.
  auto issue_chunk = [&](int c, int buf) {
    unsigned voff  = (unsigned)((nbase + c * 64 + lane) * 16);
    unsigned laddr = ldsBase + (unsigned)(buf * 1024 + lane * 16);
    asm volatile(
        "global_load_async_to_lds_b128 %0, %1, %2 offset:0\n\t"
        "global_load_async_to_lds_b128 %0, %1, %2 offset:512"
        :
        : "v"(laddr), "v"(voff), "s"(pbase)
        : "memory");
  };

  float acc[30];
#pragma unroll
  for (int t = 0; t < 30; ++t) acc[t] = 0.0f;

  issue_chunk(0, 0);

  for (int c = 0; c < nChunks; ++c) {
    const int buf = c & 1;
    if (c + 1 < nChunks) {
      issue_chunk(c + 1, buf ^ 1);
      asm volatile("s_wait_asynccnt 0x2" ::: "memory");  // current tile landed
    } else {
      asm volatile("s_wait_asynccnt 0x0" ::: "memory");
    }

    // Rolled on purpose: one iteration already exposes a WMMA + 8 independent
    // ~80-op chains; unrolling 4x pushed the kernel past 256 VGPRs
    // (s_set_vgpr_msb churn) with no additional useful ILP.
#pragma unroll 1
    for (int s = 0; s < 4; ++s) {
      // A matrix: 16 input points, rows M = n. Lanes 0-15 carry K0,K1;
      // lanes 16-31 carry K2,K3 (per 32-bit 16x4 A layout). Branchless.
      float4 pa = tile4[buf][s * 16 + col];
      float n2 = pa.x * pa.x + pa.y * pa.y + pa.z * pa.z;
      v2f Am;
      Am.x = hi ? pa.z : pa.x;
      Am.y = hi ? n2   : pa.y;

      v8f D = __builtin_amdgcn_wmma_f32_16x16x4_f32(
          false, Am, false, Bm, (short)0, Cm, false, false);

#pragma unroll
      for (int j = 0; j < 8; ++j) {
        const int nl = s * 16 + j + (hi ? 8 : 0); // row n this D[j] belongs to
        float4 p = tile4[buf][nl];                // broadcast across 16 lanes
        float r2 = __builtin_fabsf(D[j]);         // |.| guards cancellation;
        float r  = __builtin_amdgcn_sqrtf(r2);    // free src modifier on sqrt
        float x = Xa - p.x, y = Ya - p.y, z = Za - p.z;
        float fw = p.w;

        // One transcendental: u = 2^(-r*log2e/12) = exp(-r/12); k_n = u^(12/n)*f
        float u  = __builtin_amdgcn_exp2f(r * -0.12022458674074695f);
        float u2 = u * u;
        float u3 = u2 * u;     // exp(-r/4)
        float u4 = u2 * u2;    // exp(-r/3)
        float u6 = u3 * u3;    // exp(-r/2)
        float u12 = u6 * u6;   // exp(-r)
        float k1 = u12 * fw, k2 = u6 * fw, k3 = u4 * fw, k4 = u3 * fw;

        float x2 = x * x, y2 = y * y, z2 = z * z;
        float xy = x * y, xz = x * z, yz = y * z;
        float sxy = x2 + y2;
        float rm6 = r - 6.0f, rm12 = r - 12.0f;
        float p4 = 80.0f + (r - 20.0f) * r;

        acc[0]  += k1;
        acc[1]  += k2 * (r - 2.0f);
        acc[2]  -= k2 * x;
        acc[3]  += k2 * z;
        acc[4]  -= k2 * y;
        acc[5]  += k3 * (27.0f + 2.0f * (r - 9.0f) * r);
        acc[6]  += k3 * rm6 * x;
        acc[7]  -= k3 * rm6 * z;
        acc[8]  += k3 * rm6 * y;
        acc[9]  += k3 * xy;
        acc[10] -= k3 * xz;
        acc[11] += k3 * (2.0f * r2 - 3.0f * sxy);
        acc[12] -= k3 * yz;
        acc[13] += k3 * (y2 - x2);
        acc[14] += k4 * (192.0f - rm12 * rm12 * r);
        acc[15] -= k4 * p4 * x;
        acc[16] += k4 * p4 * z;
        acc[17] -= k4 * p4 * y;
        acc[18] -= k4 * rm12 * xy;
        acc[19] += k4 * rm12 * xz;
        acc[20] -= k4 * rm12 * (2.0f * r2 - 3.0f * sxy);
        acc[21] += k4 * rm12 * yz;
        acc[22] += k4 * rm12 * (x2 - y2);
        acc[23] += k4 * x * (x2 - 3.0f * y2);
        acc[24] += k4 * xy * z;
        acc[25] += k4 * x * (5.0f * sxy - 4.0f * r2);
        acc[26] += k4 * (2.0f * z2 * z - 3.0f * sxy * z);
        acc[27] += k4 * y * (5.0f * sxy - 4.0f * r2);
        acc[28] += k4 * (y2 - x2) * z;
        acc[29] += k4 * y * (3.0f * x2 - y2);
      }
    }
  }

  // Normalization constants (compiler constant-folds the sqrtf of literals).
  const float c8pi = 25.132741228718345f;            // 8*pi
  const float s2c  = 1.0f / sqrtf(c8pi * 32.0f);
  const float e3   = 6561.0f;                         // 81^2
  const float s3p  = 1.0f / sqrtf(c8pi * e3 * 2.0f);
  const float e4   = 262144.0f;                       // 512^2
  const float s45  = 1.0f / sqrtf(c8pi * e4 * 5.0f);
  const float s43  = 1.0f / sqrtf(c8pi * e4 * 3.0f);
  const float s412 = 1.0f / sqrtf(c8pi * e4 * 12.0f);
  const float s472 = 1.0f / sqrtf(c8pi * e4 * 72.0f);
  const float s4120 = 1.0f / sqrtf(c8pi * e4 * 120.0f);
  const float sc[30] = {
      1.0f / sqrtf(c8pi),
      s2c, s2c, s2c, s2c,
      1.0f / sqrtf(c8pi * e3 * 3.0f),
      s3p, s3p, s3p, s3p, s3p,
      1.0f / sqrtf(c8pi * e3 * 6.0f),
      s3p, s3p,
      1.0f / sqrtf(c8pi * e4 * 9.0f),
      s45, s45, s45,
      s43, s43,
      1.0f / sqrtf(c8pi * e4 * 36.0f),
      s43, s412, s472, s43, s4120,
      1.0f / sqrtf(c8pi * e4 * 180.0f),
      s4120, s412, s472};

  // Merge the two half-lanes sharing the same output point a.
#pragma unroll
  for (int t = 0; t < 30; ++t) acc[t] += __shfl_xor(acc[t], 16);

  if (!hi) {
    float* op = out + ((size_t)(b * A + a)) * 30;
#pragma unroll
    for (int t = 0; t < 30; ++t) atomicAdd(&op[t], acc[t] * sc[t]);
  }
}

// ---------------------------------------------------------------------------
extern "C" void kernel_launch(void* const* d_in, const int* in_sizes, int n_in,
                              void* d_out, int out_size, void* d_ws, size_t ws_size,
                              hipStream_t stream) {
  const float* f          = (const float*)d_in[0];
  const float* coords     = (const float*)d_in[1];
  const float* out_coords = (const float*)d_in[2];
  float* out = (float*)d_out;

  const int B = 2;
  const int N = in_sizes[0] / B;         // 4096
  const int A = in_sizes[2] / (3 * B);   // 2048

  float4* pts = (float4*)d_ws;           // needs B*N*16 = 128 KB of workspace
  const int total = B * N;

  // Partial sums are atomically accumulated: zero the output each call.
  hipMemsetAsync(d_out, 0, (size_t)out_size * sizeof(float), stream);

  pack_pts_kernel<<<(total + 255) / 256, 256, 0, stream>>>(f, coords, pts, total);

  const int NSPLIT = 8;                  // 2048 single-wave blocks total
  dim3 grid((B * A) / 16, NSPLIT);
  hydrogen_pool_kernel<<<grid, 32, 0, stream>>>(out_coords, pts, out,
                                                A, N, N / NSPLIT);
}